// SparseGraphOperations_63393717289136
// MI455X (gfx1250) — compile-verified
//
#include <hip/hip_runtime.h>
#include <math.h>

// ---------------------------------------------------------------------------
// SparseGraphOperations on MI455X (gfx1250, wave32)
// Reference returns ONLY _mha(x); the sparse-adjacency branch is dead code.
// All matmuls run on V_WMMA_F32_16X16X4_F32 (fp32 exactness vs. reference).
// Attention stages K/V into LDS via async global->LDS copies (ASYNCcnt path).
// ---------------------------------------------------------------------------

typedef float v2f __attribute__((ext_vector_type(2)));
typedef float v4f __attribute__((ext_vector_type(4)));
typedef float v8f __attribute__((ext_vector_type(8)));
typedef int   v4i __attribute__((vector_size(16)));   // matches builtin param

#define AS1 __attribute__((address_space(1)))
#define AS3 __attribute__((address_space(3)))

static constexpr int D  = 128;   // d_model
static constexpr int N  = 512;   // nodes
static constexpr int B  = 4;     // batch
static constexpr int H  = 8;     // heads
static constexpr int HD = 16;    // head dim
static constexpr int PER_MAT = B * H * N * HD;  // 262144 floats per Q/K/V/O buffer

__device__ __forceinline__ v8f wmma4(v2f a, v2f b, v8f c) {
  // (neg_a, A, neg_b, B, c_mod, C, reuse_a, reuse_b)
  return __builtin_amdgcn_wmma_f32_16x16x4_f32(false, a, false, b, (short)0, c,
                                               false, false);
}

__device__ __forceinline__ v2f ld2(const float* __restrict__ p) {
  v2f r;
  r[0] = p[0];
  r[1] = p[1];
  return r;
}

// Async copy of 16 bytes global -> LDS (falls back to a plain copy if the
// builtin is unavailable on this toolchain).
__device__ __forceinline__ void cp_async16(const float* gsrc, float* lds_dst) {
#if __has_builtin(__builtin_amdgcn_global_load_async_to_lds_b128)
  __builtin_amdgcn_global_load_async_to_lds_b128(
      (AS1 v4i*)(unsigned long long)gsrc,
      (AS3 v4i*)(unsigned int)(unsigned long long)lds_dst,
      /*offset=*/0, /*cpol=*/0);
#else
  *(v4f*)lds_dst = *(const v4f*)gsrc;
#endif
}

__device__ __forceinline__ void cp_async_wait() {
#if __has_builtin(__builtin_amdgcn_global_load_async_to_lds_b128)
#if __has_builtin(__builtin_amdgcn_s_wait_asynccnt)
  __builtin_amdgcn_s_wait_asynccnt(0);
#else
  asm volatile("s_wait_asynccnt 0" ::: "memory");
#endif
#endif
}

// ---------------------------------------------------------------------------
// Kernel 1: qkv = x @ in_proj_w^T + in_proj_b, scattered to per-(b,h) layout
//   x: [2048,128]  W: [384,128]  ->  q/k/v ws: [B*H][N][16]
// 3072 output tiles of 16x16; 8 waves per block, one tile per wave.
// ---------------------------------------------------------------------------
__global__ __launch_bounds__(256) void qkv_kernel(
    const float* __restrict__ x, const float* __restrict__ W,
    const float* __restrict__ bias, float* __restrict__ qws,
    float* __restrict__ kws, float* __restrict__ vws) {
  const int lane = threadIdx.x & 31;
  const int wave = threadIdx.x >> 5;
  const int half = lane >> 4;   // A/B frag K-group, C/D row-group
  const int lid  = lane & 15;

  const int tile = blockIdx.x * 8 + wave;   // 0..3071
  const int tm = tile / 24;                 // 128 row tiles over 2048
  const int tn = tile % 24;                 // 24 col tiles over 384
  const int m0 = tm * 16, n0 = tn * 16;

  v8f acc = {};
  const float* pa = x + (m0 + lid) * D + 2 * half;
  const float* pb = W + (n0 + lid) * D + 2 * half;
#pragma unroll
  for (int k0 = 0; k0 < D; k0 += 4)
    acc = wmma4(ld2(pa + k0), ld2(pb + k0), acc);

  const float bv  = bias[n0 + lid];
  const int part  = n0 / D;           // 0=q 1=k 2=v (uniform per tile)
  const int h     = (n0 % D) / HD;    // head (uniform: 16-col tiles align)
  const int b_idx = m0 / N;           // batch (uniform: 16 | 512)
  float* dst = (part == 0) ? qws : (part == 1) ? kws : vws;
  float* base = dst + ((b_idx * H + h) * N) * HD;
#pragma unroll
  for (int r = 0; r < 8; ++r) {
    const int node = (m0 + r + 8 * half) & (N - 1);   // C/D row M = r+8*half
    base[node * HD + lid] = acc[r] + bv;
  }
}

// ---------------------------------------------------------------------------
// Kernel 2: per-(b,h) flash attention. One wave owns 16 query rows.
//   grid = 32 (b,h) * 4 query chunks; 8 waves/block -> 16 rows each.
// K and V for the (b,h) are staged into LDS once per block with async
// global->LDS copies; S = Q K^T via WMMA; online softmax; P routed through a
// per-wave LDS tile to build A-fragments for O += P V (also WMMA).
// ---------------------------------------------------------------------------
__global__ __launch_bounds__(256) void attn_kernel(
    const float* __restrict__ qws, const float* __restrict__ kws,
    const float* __restrict__ vws, float* __restrict__ ows) {
  const int lane = threadIdx.x & 31;
  const int wave = threadIdx.x >> 5;
  const int half = lane >> 4;
  const int lid  = lane & 15;

  const int bh    = blockIdx.x >> 2;  // 0..31
  const int chunk = blockIdx.x & 3;   // 0..3
  const int b = bh >> 3, h = bh & 7;
  const int q0 = chunk * 128 + wave * 16;

  const float* Q = qws + bh * N * HD;
  const float* K = kws + bh * N * HD;
  const float* V = vws + bh * N * HD;

  __shared__ float kbuf[N * HD];        // 32 KB
  __shared__ float vbuf[N * HD];        // 32 KB
  __shared__ float pstage[8][16][16];   // private 16x16 P tile per wave (8 KB)
  float(*P)[16] = pstage[wave];

  // Stage K and V into LDS: 8192 floats each, 256 threads x 8 b128 ops each.
#pragma unroll
  for (int i = 0; i < 8; ++i) {
    const int idx = (i * 256 + threadIdx.x) * 4;
    cp_async16(K + idx, kbuf + idx);
    cp_async16(V + idx, vbuf + idx);
  }
  cp_async_wait();
  __syncthreads();

  // Q tile as A-fragments (K=16 -> 4 wmma k-steps)
  v2f aq[4];
#pragma unroll
  for (int ks = 0; ks < 4; ++ks)
    aq[ks] = ld2(Q + (q0 + lid) * HD + ks * 4 + 2 * half);

  v8f acc = {};
  float m[8], l[8];
#pragma unroll
  for (int r = 0; r < 8; ++r) { m[r] = -__builtin_inff(); l[r] = 0.f; }

  for (int n0 = 0; n0 < N; n0 += 16) {
    // S tile = Q * K^T  (B-frag: lane holds key row n0+lid, K idx 2*half+j)
    v8f s = {};
#pragma unroll
    for (int ks = 0; ks < 4; ++ks) {
      const float* pk = kbuf + (n0 + lid) * HD + ks * 4 + 2 * half;
      v2f bk;
      bk[0] = pk[0];
      bk[1] = pk[1];
      s = wmma4(aq[ks], bk, s);
    }

    // online softmax; row = r + 8*half, lanes of a 16-group share a row set
#pragma unroll
    for (int r = 0; r < 8; ++r) {
      const float sv = s[r] * 0.25f;   // 1/sqrt(16)
      float rmax = sv;
#pragma unroll
      for (int off = 1; off < 16; off <<= 1)
        rmax = fmaxf(rmax, __shfl_xor(rmax, off, 32));
      const float mnew = fmaxf(m[r], rmax);
      const float sc = __expf(m[r] - mnew);
      const float pv = __expf(sv - mnew);
      float rs = pv;
#pragma unroll
      for (int off = 1; off < 16; off <<= 1)
        rs += __shfl_xor(rs, off, 32);
      l[r] = l[r] * sc + rs;
      m[r] = mnew;
      acc[r] = acc[r] * sc;
      P[r + 8 * half][lid] = pv;   // stage P in [row][col] for transpose read
    }
    // same-wave DS ops are in-order, but fence data + stop compiler motion
    asm volatile("s_wait_dscnt 0" ::: "memory");

    // O += P * V : A-frag of P from LDS, B-frag of V from LDS
#pragma unroll
    for (int ks = 0; ks < 4; ++ks) {
      const int kk = ks * 4 + 2 * half;
      v2f ap;
      ap[0] = P[lid][kk];
      ap[1] = P[lid][kk + 1];
      v2f bv;
      bv[0] = vbuf[(n0 + kk) * HD + lid];
      bv[1] = vbuf[(n0 + kk + 1) * HD + lid];
      acc = wmma4(ap, bv, acc);
    }
  }

  // normalize and scatter to [b][node][h*16+c] for the output projection
#pragma unroll
  for (int r = 0; r < 8; ++r) {
    const int node = q0 + r + 8 * half;
    ows[(b * N + node) * D + h * HD + lid] = acc[r] / l[r];
  }
}

// ---------------------------------------------------------------------------
// Kernel 3: out = o @ out_proj_w^T + out_proj_b   (2048x128x128)
// ---------------------------------------------------------------------------
__global__ __launch_bounds__(256) void out_kernel(
    const float* __restrict__ o, const float* __restrict__ W,
    const float* __restrict__ bias, float* __restrict__ out) {
  const int lane = threadIdx.x & 31;
  const int wave = threadIdx.x >> 5;
  const int half = lane >> 4;
  const int lid  = lane & 15;

  const int tile = blockIdx.x * 8 + wave;  // 0..1023
  const int tm = tile >> 3, tn = tile & 7;
  const int m0 = tm * 16, n0 = tn * 16;

  v8f acc = {};
  const float* pa = o + (m0 + lid) * D + 2 * half;
  const float* pb = W + (n0 + lid) * D + 2 * half;
#pragma unroll
  for (int k0 = 0; k0 < D; k0 += 4)
    acc = wmma4(ld2(pa + k0), ld2(pb + k0), acc);

  const float bv = bias[n0 + lid];
#pragma unroll
  for (int r = 0; r < 8; ++r)
    out[(m0 + r + 8 * half) * D + n0 + lid] = acc[r] + bv;
}

// ---------------------------------------------------------------------------
extern "C" void kernel_launch(void* const* d_in, const int* in_sizes, int n_in,
                              void* d_out, int out_size, void* d_ws,
                              size_t ws_size, hipStream_t stream) {
  // setup_inputs order: x, adjacency, W1, b1, W2, b2,
  //                     in_proj_w, in_proj_b, out_proj_w, out_proj_b
  const float* x          = (const float*)d_in[0];
  const float* in_proj_w  = (const float*)d_in[6];
  const float* in_proj_b  = (const float*)d_in[7];
  const float* out_proj_w = (const float*)d_in[8];
  const float* out_proj_b = (const float*)d_in[9];
  float* out = (float*)d_out;

  float* qws = (float*)d_ws;           // [B*H][N][16]
  float* kws = qws + PER_MAT;
  float* vws = kws + PER_MAT;
  float* ows = vws + PER_MAT;          // [2048][128]

  qkv_kernel<<<384, 256, 0, stream>>>(x, in_proj_w, in_proj_b, qws, kws, vws);
  attn_kernel<<<128, 256, 0, stream>>>(qws, kws, vws, ows);
  out_kernel<<<128, 256, 0, stream>>>(ows, out_proj_w, out_proj_b, out);
}